// MSDeformAttn_k_67370857005435
// MI455X (gfx1250) — compile-verified
//
#include <hip/hip_runtime.h>
#include <hip/hip_bf16.h>

typedef float v2f __attribute__((ext_vector_type(2)));
typedef float v8f __attribute__((ext_vector_type(8)));

// ---------------------------------------------------------------------------
// GEMM:  Out[r*rowStride + c*colMul + colOfs] = (A[M x 256] @ W[256 x 256])[r,c] + bias[c]
// Block = 256 threads (8 waves) -> 64-row x 128-col tile; grid.y = 2 covers 256 cols.
// Each wave owns 16 cols and 4 M-subtiles: every B fragment feeds 4 WMMAs.
// V_WMMA_F32_16X16X4_F32 keeps everything fp32-exact.
// ---------------------------------------------------------------------------
__global__ __launch_bounds__(256) void gemm256_wmma(
    const float* __restrict__ A, const float* __restrict__ W,
    const float* __restrict__ bias, float* __restrict__ Out, int Mrows,
    int rowStride, int colMul, int colOfs)
{
    __shared__ float As[64 * 260];   // stride 260 (1040B, 16B-aligned rows, bank-spread)

    const int tid     = threadIdx.x;
    const int rowBase = blockIdx.x * 64;
    const int waveId  = tid >> 5;
    const int lane    = tid & 31;
    const int colBase = blockIdx.y * 128 + waveId * 16;

    // Cooperative float4 staging of the 64x256 A tile (zero-pad past Mrows).
    for (int i = tid; i < 64 * 64; i += 256) {
        const int r  = i >> 6;
        const int c4 = (i & 63) << 2;
        const int gr = rowBase + r;
        float4 v = make_float4(0.f, 0.f, 0.f, 0.f);
        if (gr < Mrows) v = *(const float4*)(A + (size_t)gr * 256 + c4);
        *(float4*)(&As[r * 260 + c4]) = v;
    }
    __syncthreads();

    // Fragment addressing per CDNA5 ISA layouts:
    //  A (16x4 f32): lanes 0-15 -> K={0,1}, lanes 16-31 -> K={2,3}; M = lane&15
    //  B (4x16 f32): lanes 0-15 -> K={0,1}, lanes 16-31 -> K={2,3}; N = lane&15
    const int arow  = lane & 15;
    const int khalf = (lane >> 4) << 1;          // 0 or 2
    const int ncol  = colBase + (lane & 15);

    v8f acc0 = {}, acc1 = {}, acc2 = {}, acc3 = {};
#pragma unroll 4
    for (int kk = 0; kk < 256; kk += 4) {
        v2f b;
        b.x = W[(size_t)(kk + khalf)     * 256 + ncol];
        b.y = W[(size_t)(kk + khalf + 1) * 256 + ncol];
        const int abase = arow * 260 + kk + khalf;
        v2f a0, a1, a2, a3;
        a0.x = As[          abase];  a0.y = As[          abase + 1];
        a1.x = As[16*260 + abase];   a1.y = As[16*260 + abase + 1];
        a2.x = As[32*260 + abase];   a2.y = As[32*260 + abase + 1];
        a3.x = As[48*260 + abase];   a3.y = As[48*260 + abase + 1];
        acc0 = __builtin_amdgcn_wmma_f32_16x16x4_f32(false, a0, false, b, (short)0, acc0, false, false);
        acc1 = __builtin_amdgcn_wmma_f32_16x16x4_f32(false, a1, false, b, (short)0, acc1, false, false);
        acc2 = __builtin_amdgcn_wmma_f32_16x16x4_f32(false, a2, false, b, (short)0, acc2, false, false);
        acc3 = __builtin_amdgcn_wmma_f32_16x16x4_f32(false, a3, false, b, (short)0, acc3, false, false);
    }

    // D layout: VGPR i -> M = i (lanes 0-15) or 8+i (lanes 16-31), N = lane&15
    const float  bv     = bias[ncol];
    const int    subRow = (lane >> 4) << 3;
    const size_t step   = (size_t)rowStride;
    const size_t colOff = (size_t)ncol * colMul + colOfs;

    if (rowBase + 64 <= Mrows) {           // fast path: no row checks
        float* p0 = Out + (size_t)(rowBase + subRow) * step + colOff;
        float* p1 = p0 + 16 * step;
        float* p2 = p0 + 32 * step;
        float* p3 = p0 + 48 * step;
#pragma unroll
        for (int i = 0; i < 8; ++i) {
            *p0 = acc0[i] + bv;  p0 += step;
            *p1 = acc1[i] + bv;  p1 += step;
            *p2 = acc2[i] + bv;  p2 += step;
            *p3 = acc3[i] + bv;  p3 += step;
        }
    } else {                               // tail tile
        v8f accs[4] = {acc0, acc1, acc2, acc3};
#pragma unroll
        for (int t = 0; t < 4; ++t)
#pragma unroll
            for (int i = 0; i < 8; ++i) {
                const int r = rowBase + t * 16 + subRow + i;
                if (r < Mrows) Out[(size_t)r * step + colOff] = accs[t][i] + bv;
            }
    }
}

// ---------------------------------------------------------------------------
// Fused bilinear sampling + key attention + value aggregation.
// One wave per (n, q, head); lane = channel d. value/key are interleaved as
// [pos, head, ch, {v,k}] so each corner gather is ONE float2 load per lane
// (a single coalesced 256B transaction per wave). 8 heads of a query share a
// block so their gathers touch adjacent chunks of the same rows (L2-resident:
// vk buffer = 91 MB < 192 MB L2).
// ---------------------------------------------------------------------------
__global__ __launch_bounds__(256) void msda_sample_attn(
    const float* __restrict__ vk,    const float* __restrict__ qproj,
    const float* __restrict__ off,   const float* __restrict__ refp,
    const int* __restrict__ shapes,  const int* __restrict__ starts,
    float* __restrict__ inter, int Lq, int Len_in)
{
    const int  lane = threadIdx.x & 31;            // channel d
    const int  m    = threadIdx.x >> 5;            // head
    const long nq   = blockIdx.x;                  // n*Lq + q
    const int  n    = (int)(nq / Lq);
    const int  d    = lane;

    const float qd   = qproj[nq * 256 + m * 32 + d];
    const float offv = off[nq * 256 + m * 32 + lane];  // 32 offsets, one per lane

    float sv[16];
    float logit[16];
    const float inv_sqrt_dh = 0.17677669529663687f;    // 1/sqrt(32)

#pragma unroll
    for (int l = 0; l < 4; ++l) {
        const int   H  = shapes[l * 2 + 0];
        const int   W  = shapes[l * 2 + 1];
        const float fH = (float)H, fW = (float)W;
        const float rx = refp[(nq * 4 + l) * 2 + 0];
        const float ry = refp[(nq * 4 + l) * 2 + 1];
        const long  lvlBase = (long)n * Len_in + starts[l];
#pragma unroll
        for (int p = 0; p < 4; ++p) {
            const float ox = __shfl(offv, l * 8 + p * 2 + 0, 32);
            const float oy = __shfl(offv, l * 8 + p * 2 + 1, 32);
            // grid_sample(align_corners=False): x = (rx + ox/W)*W - 0.5
            //                                     = rx*W + ox - 0.5  (divide-free)
            const float x = fmaf(rx, fW, ox) - 0.5f;
            const float y = fmaf(ry, fH, oy) - 0.5f;
            const float x0f = floorf(x), y0f = floorf(y);
            const float wx = x - x0f, wy = y - y0f;
            const int   ix0 = (int)x0f, iy0 = (int)y0f;

            float vacc = 0.0f, kacc = 0.0f;
#pragma unroll
            for (int cy = 0; cy < 2; ++cy) {
                const int iy = iy0 + cy;
                if (iy < 0 || iy >= H) continue;
                const float wyc = cy ? wy : (1.0f - wy);
#pragma unroll
                for (int cx = 0; cx < 2; ++cx) {
                    const int ix = ix0 + cx;
                    if (ix < 0 || ix >= W) continue;
                    const float wgt = wyc * (cx ? wx : (1.0f - wx));
                    const long base =
                        (lvlBase + (long)iy * W + ix) * 512 + m * 64 + d * 2;
                    const float2 vk2 = *(const float2*)(vk + base); // 1x b64/lane
                    vacc += wgt * vk2.x;
                    kacc += wgt * vk2.y;
                }
            }
            const int kidx = l * 4 + p;
            sv[kidx] = vacc;
            // q . k over 32 channels: 5-step butterfly, all lanes get the sum
            float r = qd * kacc;
#pragma unroll
            for (int s = 16; s >= 1; s >>= 1) r += __shfl_xor(r, s, 32);
            logit[kidx] = r * inv_sqrt_dh;
        }
    }

    // Softmax over 16 keys (identical in all lanes) + value aggregation.
    float mx = logit[0];
#pragma unroll
    for (int k = 1; k < 16; ++k) mx = fmaxf(mx, logit[k]);
    float ssum = 0.0f;
    float e[16];
#pragma unroll
    for (int k = 0; k < 16; ++k) { const float t = __expf(logit[k] - mx); e[k] = t; ssum += t; }
    float outd = 0.0f;
#pragma unroll
    for (int k = 0; k < 16; ++k) outd += e[k] * sv[k];
    inter[nq * 256 + m * 32 + d] = outd / ssum;
}

// ---------------------------------------------------------------------------
extern "C" void kernel_launch(void* const* d_in, const int* in_sizes, int n_in,
                              void* d_out, int out_size, void* d_ws, size_t ws_size,
                              hipStream_t stream)
{
    const float* query  = (const float*)d_in[0];
    const float* refp   = (const float*)d_in[1];
    const float* inflat = (const float*)d_in[2];
    const int*   shapes = (const int*)d_in[3];
    const int*   starts = (const int*)d_in[4];
    const float* W_off  = (const float*)d_in[5];
    const float* b_off  = (const float*)d_in[6];
    const float* W_val  = (const float*)d_in[7];
    const float* b_val  = (const float*)d_in[8];
    const float* W_key  = (const float*)d_in[9];
    const float* b_key  = (const float*)d_in[10];
    const float* W_q    = (const float*)d_in[11];
    const float* b_q    = (const float*)d_in[12];
    const float* W_out  = (const float*)d_in[13];
    const float* b_out  = (const float*)d_in[14];
    float* out = (float*)d_out;

    const int N      = 2;
    const int Lq     = in_sizes[0] / (N * 256);   // 10000
    const int Len_in = in_sizes[2] / (N * 256);   // 22223
    const int MrowsV = N * Len_in;                // 44446
    const int MrowsQ = N * Lq;                    // 20000

    float* ws = (float*)d_ws;
    size_t o = 0;
    float* vkbuf = ws + o;  o += (size_t)MrowsV * 512;   // value/key interleaved
    float* offb  = ws + o;  o += (size_t)MrowsQ * 256;
    float* qproj = ws + o;  o += (size_t)MrowsQ * 256;
    float* inter = ws + o;  // + MrowsQ*256 ; total ~152.5 MB

    const dim3 blk(256);
    const dim3 gV((MrowsV + 63) / 64, 2);
    const dim3 gQ((MrowsQ + 63) / 64, 2);

    // Projections (value/key written interleaved: rowStride=512, colMul=2)
    gemm256_wmma<<<gV, blk, 0, stream>>>(inflat, W_val, b_val, vkbuf, MrowsV, 512, 2, 0);
    gemm256_wmma<<<gV, blk, 0, stream>>>(inflat, W_key, b_key, vkbuf, MrowsV, 512, 2, 1);
    gemm256_wmma<<<gQ, blk, 0, stream>>>(query,  W_off, b_off, offb,  MrowsQ, 256, 1, 0);
    gemm256_wmma<<<gQ, blk, 0, stream>>>(query,  W_q,   b_q,   qproj, MrowsQ, 256, 1, 0);

    msda_sample_attn<<<dim3(MrowsQ), blk, 0, stream>>>(
        vkbuf, qproj, offb, refp, shapes, starts, inter, Lq, Len_in);

    gemm256_wmma<<<gQ, blk, 0, stream>>>(inter, W_out, b_out, out, MrowsQ, 256, 1, 0);
}